// FusedInplaceRotaryPositionEncoding2D_53704271069549
// MI455X (gfx1250) — compile-verified
//
#include <hip/hip_runtime.h>
#include <stdint.h>

// Problem constants (from reference): B=8, H=16, N=64*64=4096, C=128, R=32
#define CC 128
#define RR 32
#define NN 4096

#define THREADS 256
#define NBLOCKS 2048
#define ROWS_PER_ITER 16   // THREADS / 16 threads-per-row

// LDS layout: [0,16K) row_cs, [16K,32K) col_cs, [32K,48K) idx
#define LDS_BYTES 49152

typedef float v4f __attribute__((ext_vector_type(4)));

__device__ __forceinline__ void async_load_b128_to_lds(const void* gsrc, void* ldst) {
#if defined(__HIP_DEVICE_COMPILE__)
  // Low 32 bits of a flat pointer into the LDS aperture == LDS byte offset (ISA 10.2).
  uint32_t loff  = (uint32_t)(uintptr_t)ldst;
  uint64_t gaddr = (uint64_t)(uintptr_t)gsrc;
  asm volatile("global_load_async_to_lds_b128 %0, %1, off"
               :: "v"(loff), "v"(gaddr)
               : "memory");
#endif
}

__device__ __forceinline__ void wait_asynccnt0() {
#if defined(__HIP_DEVICE_COMPILE__)
  asm volatile("s_wait_asynccnt 0" ::: "memory");
#endif
}

__global__ __launch_bounds__(THREADS)
void rope2d_fused_kernel(const float* __restrict__ x,
                         const int* __restrict__ idx,
                         const float* __restrict__ row_cs,
                         const float* __restrict__ col_cs,
                         float* __restrict__ out,
                         int rows_total)
{
  __shared__ __align__(16) unsigned char s_mem[LDS_BYTES];
  const float* s_cs  = (const float*)s_mem;             // [part*4096 + pos*64 + 2k] = cos, +1 = sin
  const int*   s_idx = (const int*)(s_mem + 32768);

  const int tid = threadIdx.x;

  // Stage cos/sin tables (2 x 16KB) + packed idx (16KB) into LDS with async DMA
  // loads. Branch-free: each thread owns 4 fixed 16-B chunks of each table
  // (4 * 256 threads * 16 B = 16 KB), 12 issues total, one ASYNCcnt fence.
#pragma unroll
  for (int k = 0; k < 4; ++k) {
    const int c = tid + k * THREADS;                       // chunk 0..1023
    async_load_b128_to_lds(row_cs + (size_t)c * 4, s_mem + (size_t)c * 16);
  }
#pragma unroll
  for (int k = 0; k < 4; ++k) {
    const int c = tid + k * THREADS;
    async_load_b128_to_lds(col_cs + (size_t)c * 4, s_mem + 16384 + (size_t)c * 16);
  }
#pragma unroll
  for (int k = 0; k < 4; ++k) {
    const int c = tid + k * THREADS;
    async_load_b128_to_lds(idx + (size_t)c * 4, s_mem + 32768 + (size_t)c * 16);
  }
  wait_asynccnt0();
  __syncthreads();

  const int sub   = tid & 15;        // 16 threads cooperate on one (b,h,n) row
  const int part  = sub >> 3;        // 0: row-rotation half (ch 0..63), 1: col half (ch 64..127)
  const int k4    = (sub & 7) << 2;  // starting rotary pair index within the half (0..28)
  const int rowin = tid >> 4;        // 0..15: row within this block iteration

  const int stride = (int)gridDim.x * ROWS_PER_ITER;

  for (int g = (int)blockIdx.x * ROWS_PER_ITER + rowin; g < rows_total; g += stride) {
    const int n      = g & (NN - 1);
    const int packed = s_idx[n];
    const int pos    = part ? (packed & 0xffff) : (packed >> 16);

    const float* cs = s_cs + part * 4096 + pos * 64 + 2 * k4;   // interleaved cos/sin
    const v4f cs0 = *(const v4f*)(cs);        // (cos0, sin0, cos1, sin1)
    const v4f cs1 = *(const v4f*)(cs + 4);    // (cos2, sin2, cos3, sin3)

    const size_t base = (size_t)g * CC + (size_t)(part * 64 + k4);
    // Streamed exactly once -> non-temporal (TH=NT) to keep WGP$/L2 clean.
    const v4f a = __builtin_nontemporal_load((const v4f*)(x + base));
    const v4f b = __builtin_nontemporal_load((const v4f*)(x + base + RR));

    // Pull the next tile this thread will stream into L2 ahead of time.
    __builtin_prefetch(x + base + (size_t)stride * CC, 0, 1);

    v4f ya, yb;
    ya.x = fmaf(-b.x, cs0.y, a.x * cs0.x);   // y0 = x0*cos - x1*sin
    yb.x = fmaf( b.x, cs0.x, a.x * cs0.y);   // y1 = x0*sin + x1*cos
    ya.y = fmaf(-b.y, cs0.w, a.y * cs0.z);
    yb.y = fmaf( b.y, cs0.z, a.y * cs0.w);
    ya.z = fmaf(-b.z, cs1.y, a.z * cs1.x);
    yb.z = fmaf( b.z, cs1.x, a.z * cs1.y);
    ya.w = fmaf(-b.w, cs1.w, a.w * cs1.z);
    yb.w = fmaf( b.w, cs1.z, a.w * cs1.w);

    __builtin_nontemporal_store(ya, (v4f*)(out + base));
    __builtin_nontemporal_store(yb, (v4f*)(out + base + RR));
  }
}

extern "C" void kernel_launch(void* const* d_in, const int* in_sizes, int n_in,
                              void* d_out, int out_size, void* d_ws, size_t ws_size,
                              hipStream_t stream)
{
  const float* x      = (const float*)d_in[0];
  const int*   idx    = (const int*)d_in[1];
  const float* row_cs = (const float*)d_in[2];
  const float* col_cs = (const float*)d_in[3];
  float*       out    = (float*)d_out;

  const int rows_total = in_sizes[0] / CC;   // B*H*N = 524288

  hipLaunchKernelGGL(rope2d_fused_kernel, dim3(NBLOCKS), dim3(THREADS), 0, stream,
                     x, idx, row_cs, col_cs, out, rows_total);
}